// SobelEdgeExtractor_68977174774502
// MI455X (gfx1250) — compile-verified
//
#include <hip/hip_runtime.h>
#include <math.h>

#define IMG_H 512
#define IMG_W 512
#define NIMG  32
#define TILE  32
#define HALO  34   // TILE + 2
#define HSZ   (HALO * HALO)   // 1156 = 4*256 + 132

typedef float v4f __attribute__((ext_vector_type(4)));  // native vector: OK for nontemporal builtins

// Low 32 bits of a generic pointer to LDS == wave-relative LDS byte address
// (generic LDS addr = {shared aperture hi, lds_offset[31:0]}).
__device__ __forceinline__ unsigned lds_off(const void* p) {
  return (unsigned)(unsigned long long)(uintptr_t)p;
}

__global__ void sobel_init_max(unsigned* __restrict__ gmax) {
  if (threadIdx.x < NIMG) gmax[threadIdx.x] = 0u;
}

__global__ __launch_bounds__(256)
void sobel_pass1(const float* __restrict__ img,
                 float* __restrict__ out,
                 unsigned* __restrict__ gmax) {
  __shared__ float ch[3][HSZ];    // raw channel halo tiles (async-staged)
  __shared__ float gray[HSZ];     // gray halo tile
  __shared__ float sred[256];     // max reduction

  const int t  = threadIdx.x;            // 0..255
  const int b  = blockIdx.z;             // 0..31
  const int x0 = blockIdx.x * TILE;
  const int y0 = blockIdx.y * TILE;
  const size_t HW = (size_t)IMG_H * IMG_W;
  const float* ibase = img + (size_t)b * 3 * HW;

  // Stage 3 channel halo tiles into LDS via CDNA5 async global->LDS copies.
  // OOB halo elements (SAME zero padding) get LDS zeros via ds stores.
  #pragma unroll
  for (int c = 0; c < 3; ++c) {
    const float* cbase = ibase + (size_t)c * HW;
    #pragma unroll
    for (int k = 0; k < 5; ++k) {
      const int i = t + k * 256;
      if (k < 4 || i < HSZ) {            // k<4 folds away; only last chunk is guarded
        const int r  = i / HALO;
        const int cc = i - r * HALO;
        const int gy = y0 - 1 + r;
        const int gx = x0 - 1 + cc;
        const unsigned laddr = lds_off(&ch[c][i]);
        if ((unsigned)gy < (unsigned)IMG_H && (unsigned)gx < (unsigned)IMG_W) {
          const unsigned long long ga =
              (unsigned long long)(uintptr_t)(cbase + (size_t)gy * IMG_W + gx);
          asm volatile("global_load_async_to_lds_b32 %0, %1, off"
                       :: "v"(laddr), "v"(ga)
                       : "memory");
        } else {
          ch[c][i] = 0.0f;
        }
      }
    }
  }
  asm volatile("s_wait_asynccnt 0x0" ::: "memory");
  __syncthreads();

  // Gray tile: mean over channels (zero-padded channels -> zero-padded gray, matching ref).
  #pragma unroll
  for (int k = 0; k < 5; ++k) {
    const int i = t + k * 256;
    if (k < 4 || i < HSZ)
      gray[i] = (ch[0][i] + ch[1][i] + ch[2][i]) * (1.0f / 3.0f);
  }
  __syncthreads();

  // Each thread computes 4 output pixels of the 32x32 tile.
  float lmax = 0.0f;
  float* magPlane = out + (size_t)b * 3 * HW;   // channel-0 plane: L2-resident scratch
  #pragma unroll
  for (int k = 0; k < 4; ++k) {
    const int idx = t + k * 256;
    const int ly = idx >> 5;
    const int lx = idx & 31;
    const float* g = &gray[ly * HALO + lx];
    const float a00 = g[0],        a01 = g[1],            a02 = g[2];
    const float a10 = g[HALO],                            a12 = g[HALO + 2];
    const float a20 = g[2 * HALO], a21 = g[2 * HALO + 1], a22 = g[2 * HALO + 2];
    const float gxv = (a02 + 2.0f * a12 + a22) - (a00 + 2.0f * a10 + a20);
    const float gyv = (a20 + 2.0f * a21 + a22) - (a00 + 2.0f * a01 + a02);
    const float mag = sqrtf(gxv * gxv + gyv * gyv);
    lmax = fmaxf(lmax, mag);
    magPlane[(size_t)(y0 + ly) * IMG_W + (x0 + lx)] = mag;   // RT: keep in L2 for pass2
  }

  // Block max reduction (LDS tree; wave32-agnostic), then one atomic per block.
  sred[t] = lmax;
  __syncthreads();
  #pragma unroll
  for (int s = 128; s > 0; s >>= 1) {
    if (t < s) sred[t] = fmaxf(sred[t], sred[t + s]);
    __syncthreads();
  }
  if (t == 0) atomicMax(&gmax[b], __float_as_uint(sred[0]));  // mag >= 0: bit order == float order
}

__global__ __launch_bounds__(256)
void sobel_pass2(float* __restrict__ out,
                 const unsigned* __restrict__ gmax) {
  const int b = blockIdx.y;                       // uniform per block -> scalar load
  const size_t HW = (size_t)IMG_H * IMG_W;
  const float scale = 1.0f / (__uint_as_float(gmax[b]) + 1e-6f);

  const size_t i4 = ((size_t)blockIdx.x * 256 + threadIdx.x) * 4;
  float* plane = out + (size_t)b * 3 * HW;

  // Single-use read of the magnitude plane (NT), then streaming NT stores of
  // the broadcast result so the 100 MB of output doesn't thrash L2.
  const v4f m = __builtin_nontemporal_load((const v4f*)(plane + i4));
  const v4f r = m * scale;
  __builtin_nontemporal_store(r, (v4f*)(plane + i4));           // overwrite scratch
  __builtin_nontemporal_store(r, (v4f*)(plane + HW + i4));
  __builtin_nontemporal_store(r, (v4f*)(plane + 2 * HW + i4));
}

extern "C" void kernel_launch(void* const* d_in, const int* in_sizes, int n_in,
                              void* d_out, int out_size, void* d_ws, size_t ws_size,
                              hipStream_t stream) {
  const float* img = (const float*)d_in[0];
  float* out = (float*)d_out;
  unsigned* gmax = (unsigned*)d_ws;               // 32 * 4 bytes of workspace

  sobel_init_max<<<dim3(1), dim3(32), 0, stream>>>(gmax);
  sobel_pass1<<<dim3(IMG_W / TILE, IMG_H / TILE, NIMG), dim3(256), 0, stream>>>(img, out, gmax);
  // 512*512/4 float4 per image / 256 threads = 256 blocks per image
  sobel_pass2<<<dim3((IMG_H * IMG_W) / 4 / 256, NIMG), dim3(256), 0, stream>>>(out, gmax);
}